// BayesianSTDPClassic_9182640079055
// MI455X (gfx1250) — compile-verified
//
#include <hip/hip_runtime.h>

// Problem constants (from reference)
#define TB     10
#define T_     10240
#define I_     2048
#define O_     1024
#define ITERS  (T_ / TB)   // 1024
#define TW     4           // 16-wide column tiles per wave (64 columns / wave)

typedef __attribute__((ext_vector_type(2))) float v2f;
typedef __attribute__((ext_vector_type(8))) float v8f;

// ---------------------------------------------------------------------------
// Pass 1: tos[it][o] = sum_t spk[it*TB + t][o]
// ---------------------------------------------------------------------------
__global__ void tos_kernel(const float* __restrict__ spk, float* __restrict__ tos) {
  int idx = blockIdx.x * blockDim.x + threadIdx.x;   // ITERS*O_ threads
  int it = idx >> 10;            // / O_
  int o  = idx & (O_ - 1);
  const float* p = spk + (it * TB) * O_ + o;
  float s = 0.f;
#pragma unroll
  for (int t = 0; t < TB; ++t) s += p[t * O_];
  tos[idx] = s;
}

// ---------------------------------------------------------------------------
// Pass 2: toss[it] = sum_o tos[it][o]   (one block per iteration)
// ---------------------------------------------------------------------------
__global__ void toss_kernel(const float* __restrict__ tos, float* __restrict__ toss) {
  __shared__ float red[256];
  int it = blockIdx.x;
  const float* p = tos + it * O_;
  float s = 0.f;
  for (int o = threadIdx.x; o < O_; o += 256) s += p[o];
  red[threadIdx.x] = s;
  __syncthreads();
  for (int off = 128; off > 0; off >>= 1) {
    if ((int)threadIdx.x < off) red[threadIdx.x] += red[threadIdx.x + off];
    __syncthreads();
  }
  if (threadIdx.x == 0) toss[it] = red[0];
}

// ---------------------------------------------------------------------------
// Pass 3: snk[it] = sum(N_k) + sum_{j<it} toss[j]   (serial, trivial)
// ---------------------------------------------------------------------------
__global__ void snk_kernel(const float* __restrict__ toss,
                           const float* __restrict__ Nk,
                           float* __restrict__ snk) {
  float acc = 0.f;
  for (int o = 0; o < O_; ++o) acc += Nk[o];
  for (int it = 0; it < ITERS; ++it) { snk[it] = acc; acc += toss[it]; }
}

// ---------------------------------------------------------------------------
// Bias recurrence: one thread per output neuron
// ---------------------------------------------------------------------------
__global__ void bias_kernel(const float* __restrict__ biases,
                            const float* __restrict__ tos,
                            const float* __restrict__ toss,
                            const float* __restrict__ snk,
                            float* __restrict__ bout) {
  int o = blockIdx.x * blockDim.x + threadIdx.x;
  if (o >= O_) return;
  float b = biases[o];
  for (int it = 0; it < ITERS; ++it) {
    float ts  = tos[it * O_ + o];
    float db  = (__expf(-b) * ts - 1.0f) * toss[it];
    b += db / snk[it];            // mu_b = 1 / sum(nk before update)
  }
  bout[o] = b;
}

// ---------------------------------------------------------------------------
// Main persistent-tile weight kernel.
// Each wave owns a 16 x (16*TW) tile of W in registers for all 1024 steps.
// corr = spk^T @ psp via V_WMMA_F32_16X16X4_F32, K=10 zero-padded to 12.
//
// Fragment layouts (ISA 7.12.2):
//  A 16x4 f32 : lane l (m = l&15, h = l>>4): VGPR0 = A[m][2h], VGPR1 = A[m][2h+1]
//  B 4x16 f32 : lane l (n = l&15, h = l>>4): VGPR0 = B[2h][n], VGPR1 = B[2h+1][n]
//  C/D 16x16  : lane l, VGPR v holds (M = v + 8h, N = l&15)
// ---------------------------------------------------------------------------
__global__ void __launch_bounds__(256, 1)
w_kernel(const float* __restrict__ psp, const float* __restrict__ spk,
         const float* __restrict__ W,   const float* __restrict__ Nk,
         const float* __restrict__ tos, float* __restrict__ Wout) {
  const int lane = threadIdx.x & 31;
  const int wid  = (blockIdx.x * blockDim.x + threadIdx.x) >> 5;  // 0..2047
  const int row  = wid >> 5;              // 0..63   : 16-row band of O
  const int cg   = wid & 31;              // 0..31   : 64-column band of I
  const int o0   = row * 16;
  const int i0   = cg * (16 * TW);
  const int n    = lane & 15;             // N index / A row index
  const int h    = lane >> 4;             // half-wave select
  const int mrow = o0 + 8 * h;            // first M row held by this lane's VGPRs

  // Resident weight tiles
  v8f w[TW];
#pragma unroll
  for (int t = 0; t < TW; ++t)
#pragma unroll
    for (int v = 0; v < 8; ++v)
      w[t][v] = W[(mrow + v) * I_ + i0 + 16 * t + n];

  // Per-row spike-count accumulator nk (init from N_k input)
  float nk[8];
#pragma unroll
  for (int v = 0; v < 8; ++v) nk[v] = Nk[mrow + v];

  for (int it = 0; it < ITERS; ++it) {
    // Prefetch next iteration's streams (global_prefetch_b8)
    if (it + 1 < ITERS) {
      __builtin_prefetch(spk + ((it + 1) * TB) * O_ + o0 + n, 0, 0);
      __builtin_prefetch(psp + ((it + 1) * TB) * I_ + i0 + n, 0, 0);
    }

    // A fragments: spk^T tile, 3 K-steps of 4 (K 10..11 are zero pad)
    v2f a[3];
#pragma unroll
    for (int ks = 0; ks < 3; ++ks) {
      const int k0 = 4 * ks + 2 * h;
      float ax = 0.f, ay = 0.f;
      if (k0 < TB)     ax = spk[(it * TB + k0) * O_ + o0 + n];
      if (k0 + 1 < TB) ay = spk[(it * TB + k0 + 1) * O_ + o0 + n];
      v2f av; av.x = ax; av.y = ay;
      a[ks] = av;
    }

    // Per-row tos for this lane's 8 M rows; hoisted mu = 1/nk
    float tos8[8], mu[8];
#pragma unroll
    for (int v = 0; v < 8; ++v) {
      tos8[v] = tos[it * O_ + mrow + v];
      mu[v]   = 1.0f / nk[v];           // BASE_MU / nk (before update)
    }

#pragma unroll
    for (int t = 0; t < TW; ++t) {
      // B fragments: psp tile for this 16-column group
      v2f b[3];
#pragma unroll
      for (int ks = 0; ks < 3; ++ks) {
        const int k0 = 4 * ks + 2 * h;
        float bx = 0.f, by = 0.f;
        if (k0 < TB)     bx = psp[(it * TB + k0) * I_ + i0 + 16 * t + n];
        if (k0 + 1 < TB) by = psp[(it * TB + k0 + 1) * I_ + i0 + 16 * t + n];
        v2f bv; bv.x = bx; bv.y = by;
        b[ks] = bv;
      }

      v8f corr = {};
      corr = __builtin_amdgcn_wmma_f32_16x16x4_f32(false, a[0], false, b[0],
                                                   (short)0, corr, false, false);
      corr = __builtin_amdgcn_wmma_f32_16x16x4_f32(false, a[1], false, b[1],
                                                   (short)0, corr, false, false);
      corr = __builtin_amdgcn_wmma_f32_16x16x4_f32(false, a[2], false, b[2],
                                                   (short)0, corr, false, false);

      // w += mu * (corr * C * exp(-w) - tos)
#pragma unroll
      for (int v = 0; v < 8; ++v) {
        float e  = __expf(-w[t][v]);
        float dw = corr[v] * e - tos8[v];
        w[t][v] += mu[v] * dw;
      }
    }

#pragma unroll
    for (int v = 0; v < 8; ++v) nk[v] += tos8[v];
  }

  // Final weights out
#pragma unroll
  for (int t = 0; t < TW; ++t)
#pragma unroll
    for (int v = 0; v < 8; ++v)
      Wout[(mrow + v) * I_ + i0 + 16 * t + n] = w[t][v];
}

// ---------------------------------------------------------------------------
extern "C" void kernel_launch(void* const* d_in, const int* in_sizes, int n_in,
                              void* d_out, int out_size, void* d_ws, size_t ws_size,
                              hipStream_t stream) {
  const float* psp    = (const float*)d_in[0];  // (T, I)
  const float* spk    = (const float*)d_in[1];  // (T, O)
  const float* Win    = (const float*)d_in[2];  // (O, I)
  const float* biases = (const float*)d_in[3];  // (O,)
  const float* Nk     = (const float*)d_in[4];  // (O, 1)

  float* Wout = (float*)d_out;                  // O*I weights
  float* bout = Wout + (size_t)O_ * I_;         // then O biases

  float* tos  = (float*)d_ws;                   // ITERS * O_
  float* toss = tos + (size_t)ITERS * O_;       // ITERS
  float* snk  = toss + ITERS;                   // ITERS

  // Pass 1: per-batch per-neuron spike counts
  tos_kernel<<<(ITERS * O_) / 256, 256, 0, stream>>>(spk, tos);
  // Pass 2: per-batch total spikes
  toss_kernel<<<ITERS, 256, 0, stream>>>(tos, toss);
  // Pass 3: prefix sums -> sum(nk) before each update
  snk_kernel<<<1, 1, 0, stream>>>(toss, Nk, snk);

  // Main weight scan: 64 row-bands x 32 col-groups = 2048 waves, 8 waves/block
  w_kernel<<<256, 256, 0, stream>>>(psp, spk, Win, Nk, tos, Wout);

  // Bias scan
  bias_kernel<<<(O_ + 255) / 256, 256, 0, stream>>>(biases, tos, toss, snk, bout);
}